// Distiller_53412213293408
// MI455X (gfx1250) — compile-verified
//
#include <hip/hip_runtime.h>
#include <stdint.h>

// MI455X / gfx1250 wave32 flash-attention-style kernel using fp32 WMMA and
// async global->LDS DMA (GLOBAL_LOAD_ASYNC_TO_LDS_B128, ASYNCcnt-tracked).

typedef float v2f __attribute__((ext_vector_type(2)));
typedef float v8f __attribute__((ext_vector_type(8)));

#define WMMA4(a, b, c) \
  __builtin_amdgcn_wmma_f32_16x16x4_f32(false, (a), false, (b), (short)0, (c), false, false)

#define N_TOK 1024
#define RSTR  1028   // padded LDS row stride (dwords): 1028 % 64 == 4 -> no bank conflicts
#define PSTR  17     // padded stride for P staging tile
#define USE_ASYNC_FILL 1

__global__ __launch_bounds__(256) void attn_sse_kernel(
    const float* __restrict__ fs,   // f_s [8,256,32,32]
    const float* __restrict__ ft,   // f_t [8,256,32,32]
    float* __restrict__ partial)    // [512] per-block SSE partials
{
  extern __shared__ float smem[];
  float* S2  = smem;                 // [32][1028]  f_s slab, S2[d][t] (Q and V)
  float* T2  = S2 + 32 * RSTR;       // [32][1028]  f_t slab, T2[d][t] (K, MSE target)
  float* Pst = T2 + 32 * RSTR;       // 8 waves x [16][17] staging
  float* wred = Pst + 8 * 16 * PSTR; // [8]

  const int tid  = threadIdx.x;
  const int wave = tid >> 5;
  const int lane = tid & 31;
  const int m    = lane & 15;   // row/col index inside a 16-wide fragment
  const int hh   = lane >> 4;   // lane half -> K / M-offset selector

  const int pair = blockIdx.x >> 3;   // 64 (b,h) pairs
  const int qblk = blockIdx.x & 7;    // 8 row-blocks of 128 Q rows
  const int bb   = pair >> 3;
  const int head = pair & 7;
  const size_t base = ((size_t)(bb * 256 + head * 32)) * (size_t)N_TOK;

  // ---- async DMA fill: S2[d][t] = fs[base + d*256 + t], d < 128, t < 256 (padded memcpy) ----
  {
    const float4* gs = (const float4*)(fs + base);   // 8192 float4 chunks
    const float4* gt = (const float4*)(ft + base);
#pragma unroll
    for (int it = 0; it < 32; ++it) {
      const int c = tid + it * 256;            // chunk id 0..8191
      const int d = c >> 6;                    // 64 chunks per 256-dword row
      const int o = (c & 63) << 2;             // dword offset within row
      float* ds_ = &S2[d * RSTR + o];
      float* dt_ = &T2[d * RSTR + o];
#if USE_ASYNC_FILL
      // per-lane 16B global -> LDS, no VGPR round trip; tracked by ASYNCcnt
      asm volatile("global_load_async_to_lds_b128 %0, %1, off"
                   :: "v"((uint32_t)(uintptr_t)ds_),
                      "v"((uint64_t)(uintptr_t)(gs + c))
                   : "memory");
      asm volatile("global_load_async_to_lds_b128 %0, %1, off"
                   :: "v"((uint32_t)(uintptr_t)dt_),
                      "v"((uint64_t)(uintptr_t)(gt + c))
                   : "memory");
#else
      *(float4*)ds_ = gs[c];
      *(float4*)dt_ = gt[c];
#endif
    }
#if USE_ASYNC_FILL
    asm volatile("s_wait_asynccnt 0" ::: "memory");
#endif
  }
  __syncthreads();

  // ---- per-wave 16-row Q tile ----
  const int i0 = qblk * 128 + wave * 16;

  // Q A-fragment: a.x = Q[i0+m][c] = S2[c][i0+m], c = 4*k0 + 2*hh
  v2f aQ[8];
#pragma unroll
  for (int k0 = 0; k0 < 8; ++k0) {
    const int c = 4 * k0 + 2 * hh;
    aQ[k0] = (v2f){S2[c * RSTR + i0 + m], S2[(c + 1) * RSTR + i0 + m]};
  }

  v8f O0 = {};    // out cols d=0..15   (C/D layout: M = g + 8*hh, N = m)
  v8f O1 = {};    // out cols d=16..31
  v8f L  = {};    // row sums of exp(scores); every column identical
  const v2f bOnes = (v2f){1.0f, 1.0f};
  float* pw = Pst + wave * 16 * PSTR;

  for (int j0 = 0; j0 < N_TOK; j0 += 16) {
    // scores tile = Q * K^T ; B-frag: b.x = K[j0+m][c] = T2[c][j0+m]
    v8f Sacc = {};
#pragma unroll
    for (int k0 = 0; k0 < 8; ++k0) {
      const int c = 4 * k0 + 2 * hh;
      v2f bK = (v2f){T2[c * RSTR + j0 + m], T2[(c + 1) * RSTR + j0 + m]};
      Sacc = WMMA4(aQ[k0], bK, Sacc);
    }
    // P = exp(scores) — safe without max subtraction (|score| < ~35 in fp32)
    v8f P;
#pragma unroll
    for (int g = 0; g < 8; ++g) P[g] = __expf(Sacc[g]);

    // stage P (C/D layout) -> wave-private LDS -> reload in A layout.
    // Intra-wave LDS ops are in-order; fences stop compiler reordering.
    __asm__ volatile("" ::: "memory");
#pragma unroll
    for (int g = 0; g < 8; ++g) pw[(g + 8 * hh) * PSTR + m] = P[g];
    __asm__ volatile("" ::: "memory");

#pragma unroll
    for (int k0 = 0; k0 < 4; ++k0) {
      const float* pa = &pw[m * PSTR + 4 * k0 + 2 * hh];
      v2f aP = (v2f){pa[0], pa[1]};
      const int jr = j0 + 4 * k0 + 2 * hh;                 // even -> 8B aligned
      v2f bV0 = *(const v2f*)&S2[m * RSTR + jr];           // V[jr..jr+1][m]
      v2f bV1 = *(const v2f*)&S2[(m + 16) * RSTR + jr];    // V[jr..jr+1][m+16]
      O0 = WMMA4(aP, bV0, O0);
      O1 = WMMA4(aP, bV1, O1);
      L  = WMMA4(aP, bOnes, L);   // row-sum accumulator, same C/D layout
    }
    __asm__ volatile("" ::: "memory");
  }

  // ---- normalize (pure per-lane: O and L share the C/D layout) + SSE ----
  float sse = 0.0f;
#pragma unroll
  for (int g = 0; g < 8; ++g) {
    const float inv = 1.0f / L[g];
    const int row = i0 + g + 8 * hh;           // token index of this element
    const float e0 = O0[g] * inv - T2[m * RSTR + row];
    const float e1 = O1[g] * inv - T2[(m + 16) * RSTR + row];
    sse += e0 * e0 + e1 * e1;
  }

  // wave32 reduction, then block reduction via LDS
#pragma unroll
  for (int off = 16; off > 0; off >>= 1) sse += __shfl_xor(sse, off, 32);
  if (lane == 0) wred[wave] = sse;
  __syncthreads();
  if (tid == 0) {
    float tot = 0.0f;
#pragma unroll
    for (int w = 0; w < 8; ++w) tot += wred[w];
    partial[blockIdx.x] = tot;
  }
}

__global__ __launch_bounds__(256) void reduce_kernel(
    const float* __restrict__ partial, float* __restrict__ out)
{
  __shared__ float red[256];
  float s = 0.0f;
  for (int i = threadIdx.x; i < 512; i += 256) s += partial[i];
  red[threadIdx.x] = s;
  __syncthreads();
  for (int step = 128; step > 0; step >>= 1) {
    if (threadIdx.x < step) red[threadIdx.x] += red[threadIdx.x + step];
    __syncthreads();
  }
  if (threadIdx.x == 0)
    out[0] = red[0] * (1.0f / (8.0f * 256.0f * 32.0f * 32.0f));
}

extern "C" void kernel_launch(void* const* d_in, const int* in_sizes, int n_in,
                              void* d_out, int out_size, void* d_ws, size_t ws_size,
                              hipStream_t stream) {
  const float* fs = (const float*)d_in[0];
  const float* ft = (const float*)d_in[1];
  float* out = (float*)d_out;
  float* partial = (float*)d_ws;   // 512 floats of scratch

  const size_t shmem =
      (size_t)(2 * 32 * RSTR + 8 * 16 * PSTR + 8) * sizeof(float); // ~272 KB

  // Opt in to >64KB dynamic LDS (WGP has 320 KB on CDNA5).
  (void)hipFuncSetAttribute(reinterpret_cast<const void*>(attn_sse_kernel),
                            hipFuncAttributeMaxDynamicSharedMemorySize,
                            (int)shmem);

  attn_sse_kernel<<<512, 256, shmem, stream>>>(fs, ft, partial);
  reduce_kernel<<<1, 256, 0, stream>>>(partial, out);
}